// ConvMultiHeadAttention_34883724378569
// MI455X (gfx1250) — compile-verified
//
#include <hip/hip_runtime.h>
#include <hip/hip_bf16.h>
#include <math.h>

typedef __attribute__((ext_vector_type(2))) float v2f;
typedef __attribute__((ext_vector_type(8))) float v8f;

#define NB 8          // batch
#define NH 8          // heads
#define SLEN 976      // sequence length
#define DH 64         // head dim
#define H2 61         // pooled conv length
#define DM 512        // d_model
#define LL 16         // max_len
#define QT 61         // 16-row tiles (976/16)
#define LDS_STRIDE 980  // 976 + 4 pad: 20*m mod 64 hits 16 distinct banks
#define XOUT_ELEMS (NB * SLEN * DM)  // 3,997,696

// ---------------------------------------------------------------------------
// Kernel 0: fold conv(25) + avgpool(8) into effective 32-tap stride-8 filter.
// wEff[f][c][t] = (1/8) * sum_{j=0..7, 0<=t-j<25} w_f[c][t-j]
// ---------------------------------------------------------------------------
__global__ __launch_bounds__(256) void weff_kernel(
    const float* __restrict__ wq, const float* __restrict__ wk,
    const float* __restrict__ wv, float* __restrict__ weff) {
  int tid = blockIdx.x * blockDim.x + threadIdx.x;  // 3*512*32
  if (tid >= 3 * DM * 32) return;
  int t = tid & 31;
  int c = (tid >> 5) & 511;
  int f = tid >> 14;
  const float* w = (f == 0) ? wq : (f == 1) ? wk : wv;  // (512,1,25,1)
  float acc = 0.0f;
#pragma unroll
  for (int j = 0; j < 8; ++j) {
    int tap = t - j;
    if (tap >= 0 && tap < 25) acc += w[c * 25 + tap];
  }
  weff[tid] = acc * 0.125f;
}

// ---------------------------------------------------------------------------
// Kernel 1: projection. y[b,c,h2,l] = sum_t x[b, 8*h2+t, l] * wEff[c,t]
// scattered into P[b][hh][s][d] with m = c*976 + h2*16 + l.
// ---------------------------------------------------------------------------
__global__ __launch_bounds__(256) void proj_kernel(
    const float* __restrict__ xq, const float* __restrict__ xk,
    const float* __restrict__ xv, const float* __restrict__ weff,
    float* __restrict__ Q, float* __restrict__ Kp, float* __restrict__ V) {
  long tid = (long)blockIdx.x * blockDim.x + threadIdx.x;
  const long total = 3L * NB * DM * H2 * LL;  // 11,993,088
  if (tid >= total) return;
  int l = (int)(tid & 15);
  int h2 = (int)((tid >> 4) % H2);
  long r = tid / (H2 * LL);
  int c = (int)(r & 511);
  r >>= 9;
  int b = (int)(r & 7);
  int f = (int)(r >> 3);
  const float* x = (f == 0) ? xq : (f == 1) ? xk : xv;
  const float* we = weff + ((long)f * DM + c) * 32;
  const float* xb = x + (long)b * DM * LL + (long)(8 * h2) * LL + l;
  float acc = 0.0f;
#pragma unroll
  for (int t = 0; t < 32; ++t) acc += we[t] * xb[t * LL];
  int m = c * (H2 * LL) + h2 * LL + l;
  int s = m >> 9;
  int hh = (m >> 6) & 7;
  int d = m & 63;
  float* P = (f == 0) ? Q : (f == 1) ? Kp : V;
  P[(((long)b * NH + hh) * SLEN + s) * DH + d] = acc;
}

// ---------------------------------------------------------------------------
// Kernel 2: fused attention for one (bh, qtile):
//   phase 1: scores strip (16 x 976) = QK^T/8 via V_WMMA_F32_16X16X4_F32 -> LDS
//   phase 2: row softmax in LDS (wave32 per row), write attn to d_out once
//   phase 3: out = attn(LDS) @ V via WMMA, epilogue *w_out + b_out -> d_out
// Block = 128 threads (4 waves). Grid (61, 64).
// ---------------------------------------------------------------------------
__global__ __launch_bounds__(128) void attn_fused_kernel(
    const float* __restrict__ Q, const float* __restrict__ Km,
    const float* __restrict__ V, float* __restrict__ attn,
    float* __restrict__ xout, const float* __restrict__ w_out,
    const float* __restrict__ b_out) {
  __shared__ float sA[16 * LDS_STRIDE];  // 62,720 B

  const int lane = threadIdx.x & 31;
  const int wave = threadIdx.x >> 5;  // 0..3
  const int qt = blockIdx.x;          // 0..60
  const int bh = blockIdx.y;          // 0..63
  const int hs = lane >> 4;           // half-wave select
  const int mrow = lane & 15;

  // ---- Phase 1: scores -> LDS -------------------------------------------
  // A frag: Q[qt*16 + mrow][4*i + 2*hs + {0,1}]
  const float* qrow = Q + (((long)bh * SLEN) + qt * 16 + mrow) * DH + 2 * hs;
  v2f qf[16];
#pragma unroll
  for (int i = 0; i < 16; ++i) qf[i] = *(const v2f*)(qrow + 4 * i);

  for (int kt = wave; kt < QT; kt += 4) {
    const float* krow = Km + (((long)bh * SLEN) + kt * 16 + mrow) * DH + 2 * hs;
    v8f acc0 = {};
    v8f acc1 = {};
#pragma unroll
    for (int i = 0; i < 16; i += 2) {
      v2f kf0 = *(const v2f*)(krow + 4 * i);
      v2f kf1 = *(const v2f*)(krow + 4 * i + 4);
      acc0 = __builtin_amdgcn_wmma_f32_16x16x4_f32(false, qf[i], false, kf0,
                                                   (short)0, acc0, false, false);
      acc1 = __builtin_amdgcn_wmma_f32_16x16x4_f32(false, qf[i + 1], false, kf1,
                                                   (short)0, acc1, false, false);
    }
    v8f acc = acc0 + acc1;
#pragma unroll
    for (int j = 0; j < 8; ++j) {
      // D layout: row j + 8*hs, col mrow
      sA[(j + 8 * hs) * LDS_STRIDE + kt * 16 + mrow] = acc[j] * 0.125f;
    }
  }
  __syncthreads();

  // ---- Phase 2: softmax per row, write attn to global --------------------
  float* gattn = attn + (long)bh * SLEN * SLEN + (long)(qt * 16) * SLEN;
#pragma unroll
  for (int rr = 0; rr < 4; ++rr) {
    const int r = wave * 4 + rr;  // local row 0..15
    float* pr = sA + r * LDS_STRIDE;
    float mx = -INFINITY;
    for (int i = lane; i < SLEN; i += 32) mx = fmaxf(mx, pr[i]);
#pragma unroll
    for (int o = 16; o > 0; o >>= 1) mx = fmaxf(mx, __shfl_xor(mx, o, 32));
    float sum = 0.0f;
    for (int i = lane; i < SLEN; i += 32) {
      float e = expf(pr[i] - mx);
      pr[i] = e;
      sum += e;
    }
#pragma unroll
    for (int o = 16; o > 0; o >>= 1) sum += __shfl_xor(sum, o, 32);
    const float inv = 1.0f / sum;
    float* grow = gattn + (long)r * SLEN;
    for (int i = lane; i < SLEN; i += 32) {
      float p = pr[i] * inv;
      pr[i] = p;
      grow[i] = p;
    }
  }
  __syncthreads();

  // ---- Phase 3: out = attn(LDS) @ V, epilogue ----------------------------
  const int dt = wave;  // d-tile 0..3
  const float* aL = sA + mrow * LDS_STRIDE + 2 * hs;
  const float* vcol = V + (long)bh * SLEN * DH + dt * 16 + mrow;
  v8f acc0 = {};
  v8f acc1 = {};
  for (int kc = 0; kc < 244; kc += 2) {
    const int k0 = kc * 4 + 2 * hs;
    v2f a0 = *(const v2f*)(aL + kc * 4);
    v2f a1 = *(const v2f*)(aL + kc * 4 + 4);
    v2f b0, b1;
    b0.x = vcol[(long)k0 * DH];
    b0.y = vcol[(long)(k0 + 1) * DH];
    b1.x = vcol[(long)(k0 + 4) * DH];
    b1.y = vcol[(long)(k0 + 5) * DH];
    acc0 = __builtin_amdgcn_wmma_f32_16x16x4_f32(false, a0, false, b0, (short)0,
                                                 acc0, false, false);
    acc1 = __builtin_amdgcn_wmma_f32_16x16x4_f32(false, a1, false, b1, (short)0,
                                                 acc1, false, false);
  }
  v8f acc = acc0 + acc1;
  const float wo = w_out[0];
  const float bo = b_out[0];
  const int b = bh >> 3;
  const int hh = bh & 7;
#pragma unroll
  for (int j = 0; j < 8; ++j) {
    int s = qt * 16 + j + 8 * hs;
    xout[((long)b * SLEN + s) * DM + hh * DH + dt * 16 + mrow] =
        acc[j] * wo + bo;
  }
}

// ---------------------------------------------------------------------------
extern "C" void kernel_launch(void* const* d_in, const int* in_sizes, int n_in,
                              void* d_out, int out_size, void* d_ws,
                              size_t ws_size, hipStream_t stream) {
  const float* q_in = (const float*)d_in[0];
  const float* k_in = (const float*)d_in[1];
  const float* v_in = (const float*)d_in[2];
  const float* wq = (const float*)d_in[3];
  const float* wk = (const float*)d_in[4];
  const float* wv = (const float*)d_in[5];
  const float* w_out = (const float*)d_in[6];
  const float* b_out = (const float*)d_in[7];

  float* out = (float*)d_out;
  float* xout = out;               // (8, 976, 512)
  float* attn = out + XOUT_ELEMS;  // (8, 8, 976, 976)

  // workspace layout (floats)
  float* ws = (float*)d_ws;
  float* weff = ws;      // 3*512*32 = 49152
  float* Q = ws + 65536; // 8*8*976*64 each
  float* Kp = Q + (long)NB * NH * SLEN * DH;
  float* V = Kp + (long)NB * NH * SLEN * DH;

  // 0: effective filters (conv+pool folded)
  weff_kernel<<<(3 * DM * 32 + 255) / 256, 256, 0, stream>>>(wq, wk, wv, weff);

  // 1: projections scattered to (B,H,S,D)
  {
    long total = 3L * NB * DM * H2 * LL;
    proj_kernel<<<(int)((total + 255) / 256), 256, 0, stream>>>(
        q_in, k_in, v_in, weff, Q, Kp, V);
  }

  // 2: fused scores + softmax + AV + epilogue
  dim3 tiles(QT, NB * NH);
  attn_fused_kernel<<<tiles, 128, 0, stream>>>(Q, Kp, V, attn, xout, w_out,
                                               b_out);
}